// PlainTensorProductBlock_10144712753186
// MI455X (gfx1250) — compile-verified
//
#include <hip/hip_runtime.h>
#include <hip/hip_bf16.h>

typedef float v4f __attribute__((ext_vector_type(4)));

#define WAVES_PER_BLOCK 8
#define F4_PER_ROW   32   // 128 floats / 4
#define F4_PER_MAT   96   // 3 rows * 32
#define F4_PER_WAVE 192   // x tile + y tile

// out[b,i,j,f] = x[b,i,f] * y[b,j,f]; second output = 0.75 * first.
// One wave handles one batch b; lane handles 4 consecutive features.
// Inputs staged global->LDS via CDNA5 async b128 loads (ASYNCcnt),
// outputs streamed with non-temporal b128 stores.
__global__ __launch_bounds__(256) void plain_tp_outer_kernel(
    const float* __restrict__ x,
    const float* __restrict__ y,
    float* __restrict__ out,
    int B) {
  __shared__ v4f lds[WAVES_PER_BLOCK * F4_PER_WAVE];  // 24 KB

  const int tid  = threadIdx.x;
  const int wave = tid >> 5;
  const int lane = tid & 31;
  const int b    = blockIdx.x * WAVES_PER_BLOCK + wave;
  if (b >= B) return;  // wave-uniform guard

  const v4f* __restrict__ xg = (const v4f*)x + (size_t)b * F4_PER_MAT;
  const v4f* __restrict__ yg = (const v4f*)y + (size_t)b * F4_PER_MAT;
  v4f* lw = &lds[wave * F4_PER_WAVE];

  // Stage x[b] (3 rows) and y[b] (3 rows) into this wave's LDS tile.
  // Per-lane 16B transfers; lanes are consecutive -> fully coalesced.
#pragma unroll
  for (int r = 0; r < 3; ++r) {
    unsigned lds_addr = (unsigned)(size_t)(&lw[r * F4_PER_ROW + lane]);
    const v4f* gp = xg + r * F4_PER_ROW + lane;
    asm volatile("global_load_async_to_lds_b128 %0, %1, off"
                 :: "v"(lds_addr), "v"(gp)
                 : "memory");
  }
#pragma unroll
  for (int r = 0; r < 3; ++r) {
    unsigned lds_addr = (unsigned)(size_t)(&lw[F4_PER_MAT + r * F4_PER_ROW + lane]);
    const v4f* gp = yg + r * F4_PER_ROW + lane;
    asm volatile("global_load_async_to_lds_b128 %0, %1, off"
                 :: "v"(lds_addr), "v"(gp)
                 : "memory");
  }

  // Each wave consumes only its own async loads: wait ASYNCcnt==0, no barrier.
  asm volatile("s_wait_asynccnt 0x0" ::: "memory");

  v4f xi[3], yj[3];
#pragma unroll
  for (int i = 0; i < 3; ++i) xi[i] = lw[i * F4_PER_ROW + lane];
#pragma unroll
  for (int j = 0; j < 3; ++j) yj[j] = lw[F4_PER_MAT + j * F4_PER_ROW + lane];

  // out layout (float4 units): (b*9 + i*3 + j)*32 + lane ; second output at +B*288
  v4f* __restrict__ o1 = (v4f*)out + (size_t)b * 288;
  v4f* __restrict__ o2 = (v4f*)out + (size_t)B * 288 + (size_t)b * 288;

#pragma unroll
  for (int i = 0; i < 3; ++i) {
#pragma unroll
    for (int j = 0; j < 3; ++j) {
      v4f p = xi[i] * yj[j];
      v4f q = p * 0.75f;  // NORM = 3/4
      const int o = (i * 3 + j) * F4_PER_ROW + lane;
      __builtin_nontemporal_store(p, &o1[o]);
      __builtin_nontemporal_store(q, &o2[o]);
    }
  }
}

extern "C" void kernel_launch(void* const* d_in, const int* in_sizes, int n_in,
                              void* d_out, int out_size, void* d_ws, size_t ws_size,
                              hipStream_t stream) {
  const float* x = (const float*)d_in[0];
  const float* y = (const float*)d_in[1];
  float* out = (float*)d_out;

  const int B = in_sizes[0] / (3 * 128);  // 100000
  const int blocks = (B + WAVES_PER_BLOCK - 1) / WAVES_PER_BLOCK;

  plain_tp_outer_kernel<<<blocks, 256, 0, stream>>>(x, y, out, B);
}